// PanopticPostProcessor_1236950582023
// MI455X (gfx1250) — compile-verified
//
#include <hip/hip_runtime.h>

#define BB 2
#define NC 19
#define HH 512
#define WW 1024
#define HW (HH*WW)
#define KK 200
#define KPAD 208
#define NCHUNK (KPAD/16)        // 13
#define PEAK_CAP 16384
#define FARV 1e9f
#define THING_MASK 0x0007F800u   // classes 11..18
#define CENTER_THR 0.1f
#define STUFF_AREA 4096
#define VOIDL 255

typedef __attribute__((ext_vector_type(2))) float v2f;
typedef __attribute__((ext_vector_type(8))) float v8f;

// ---------------- kernel 0: zero counters/histograms ----------------
__global__ void k_zero(int* pcnt, int* meta, int* hist, int* scnt) {
    int t = blockIdx.x * blockDim.x + threadIdx.x;
    int nh = BB * (KK + 1) * NC;
    if (t < nh) hist[t] = 0;
    if (t < BB * NC) scnt[t] = 0;
    if (t < BB) { pcnt[t] = 0; meta[t] = 0; }
}

// ---------------- kernel 1: semantic argmax over C=19 ----------------
__global__ void k_sem_argmax(const float* __restrict__ logits, int* __restrict__ sem) {
    int g = blockIdx.x * blockDim.x + threadIdx.x;   // over B*HW
    if (g >= BB * HW) return;
    int b = g / HW, p = g % HW;
    const float* base = logits + (size_t)b * NC * HW + p;
    float best = base[0]; int bi = 0;
    #pragma unroll
    for (int c = 1; c < NC; ++c) {
        float v = base[(size_t)c * HW];
        if (v > best) { best = v; bi = c; }   // strict > keeps first max (jnp.argmax)
    }
    sem[g] = bi;
}

// ---------------- kernel 2: thresholded 9x9 NMS, compact peaks ----------------
__global__ void k_nms(const float* __restrict__ heat,
                      float* __restrict__ psc, int* __restrict__ pix, int* __restrict__ pcnt) {
    __shared__ float tile[24][24];
    int b = blockIdx.z;
    int gx0 = blockIdx.x * 16 - 4, gy0 = blockIdx.y * 16 - 4;
    int tid = threadIdx.y * 16 + threadIdx.x;
    const float* hb = heat + (size_t)b * HW;
    for (int i = tid; i < 24 * 24; i += 256) {
        int ly = i / 24, lx = i % 24;
        int gy = gy0 + ly, gx = gx0 + lx;
        float v = 0.f;                         // 0 pad == -inf pad after thresholding (v>=0)
        if (gy >= 0 && gy < HH && gx >= 0 && gx < WW) {
            float h = hb[gy * WW + gx];
            v = (h > CENTER_THR) ? h : 0.f;
        }
        tile[ly][lx] = v;
    }
    __syncthreads();
    int ly = threadIdx.y + 4, lx = threadIdx.x + 4;
    float v = tile[ly][lx];
    if (v <= 0.f) return;
    float m = 0.f;
    #pragma unroll
    for (int dy = -4; dy <= 4; ++dy)
        #pragma unroll
        for (int dx = -4; dx <= 4; ++dx)
            m = fmaxf(m, tile[ly + dy][lx + dx]);
    if (v == m) {
        int pos = atomicAdd(&pcnt[b], 1);
        if (pos < PEAK_CAP) {
            psc[b * PEAK_CAP + pos] = v;
            pix[b * PEAK_CAP + pos] = (gy0 + ly) * WW + (gx0 + lx);
        }
    }
}

// ---------------- kernel 3: top-K selection + ordering + center coords ----------------
__global__ void k_topk(float* psc, int* pixbuf, const int* pcnt,
                       float* cyArr, float* cxArr, int* meta) {
    __shared__ float s_sc[256];
    __shared__ int   s_px[256];
    __shared__ int   s_pos[256];
    __shared__ int   sel[KK];
    int b = blockIdx.x, tid = threadIdx.x;
    int total = pcnt[b];
    int n = total < PEAK_CAP ? total : PEAK_CAP;
    float* sc = psc + b * PEAK_CAP;
    int*   px = pixbuf + b * PEAK_CAP;
    // iterative extraction: descending score, tie-break ascending pixel index
    for (int k = 0; k < KK; ++k) {
        float bs = 0.f; int bp = 0x7fffffff, bq = -1;
        for (int i = tid; i < n; i += 256) {
            float s = sc[i];
            if (s > bs || (s == bs && s > 0.f && px[i] < bp)) { bs = s; bp = px[i]; bq = i; }
        }
        s_sc[tid] = bs; s_px[tid] = bp; s_pos[tid] = bq;
        __syncthreads();
        for (int off = 128; off > 0; off >>= 1) {
            if (tid < off) {
                float os = s_sc[tid + off]; int op = s_px[tid + off];
                if (os > s_sc[tid] || (os == s_sc[tid] && os > 0.f && op < s_px[tid])) {
                    s_sc[tid] = os; s_px[tid] = op; s_pos[tid] = s_pos[tid + off];
                }
            }
            __syncthreads();
        }
        if (tid == 0) {
            if (s_sc[0] > 0.f) { sel[k] = s_px[0]; sc[s_pos[0]] = -1.f; }
            else sel[k] = -1;
        }
        __syncthreads();
    }
    // reference: if num_pos <= K, order by row-major pixel index (nonzero() order)
    if (total <= KK) {
        for (int ph = 0; ph < KK; ++ph) {
            int i = 2 * tid + (ph & 1);
            if (i + 1 < KK) {
                int a = sel[i], c = sel[i + 1];
                unsigned ka = (a < 0) ? 0xFFFFFFFFu : (unsigned)a;
                unsigned kc = (c < 0) ? 0xFFFFFFFFu : (unsigned)c;
                if (kc < ka) { sel[i] = c; sel[i + 1] = a; }
            }
            __syncthreads();
        }
    }
    if (tid < KPAD) {
        int p = (tid < KK) ? sel[tid] : -1;
        float cy = FARV, cx = FARV;
        if (p >= 0) { cy = (float)(p / WW); cx = (float)(p % WW); }
        cyArr[b * KPAD + tid] = cy;
        cxArr[b * KPAD + tid] = cx;
    }
    if (tid == 0) meta[b] = (n > 0) ? 1 : 0;
}

// ---------------- kernel 4: WMMA closest-center + instance assignment ----------------
// D[M][n] = ty*(-2cy) + tx*(-2cx) + 1*(cy^2+cx^2) = d2 - (ty^2+tx^2); argmin_n D == argmin_n d2.
// A (16x4 f32): lanes 0-15 hold K=0,1 = (ty,tx); lanes 16-31 hold K=2,3 = (1,0).
// B (4x16 f32): vgpr0 = rows K=0 (lanes 0-15) / K=2 (lanes 16-31); vgpr1 = K=1 / K=3.
__device__ __forceinline__ unsigned sortable_f32(float f) {
    unsigned u = __float_as_uint(f);
    return (u & 0x80000000u) ? ~u : (u | 0x80000000u);  // monotonic uint order
}

__global__ void k_closest(const float* __restrict__ offm, const int* __restrict__ sem,
                          const float* __restrict__ cyArr, const float* __restrict__ cxArr,
                          const int* __restrict__ meta, int* __restrict__ inst_out) {
    int wave = threadIdx.x >> 5;
    int lane = threadIdx.x & 31;
    int pix0 = blockIdx.x * 128 + wave * 16;   // 8 waves * 16 pixels per block
    int b = pix0 / HW;
    int p = (pix0 % HW) + (lane & 15);
    int row = p / WW, col = p % WW;
    const float* offb = offm + (size_t)b * 2 * HW;
    float ty = (float)row + offb[p];
    float tx = (float)col + offb[HW + p];
    v2f a;
    a.x = (lane < 16) ? ty : 1.0f;
    a.y = (lane < 16) ? tx : 0.0f;
    const float* cy = cyArr + b * KPAD;
    const float* cx = cxArr + b * KPAD;

    // Preload all B-matrix operands so the 13 WMMAs issue with no memory waits in between.
    v2f bmat[NCHUNK];
    #pragma unroll
    for (int ch = 0; ch < NCHUNK; ++ch) {
        int n = ch * 16 + (lane & 15);
        float yy = cy[n], xx = cx[n];
        bmat[ch].x = (lane < 16) ? (-2.0f * yy) : (yy * yy + xx * xx);
        bmat[ch].y = (lane < 16) ? (-2.0f * xx) : 0.0f;
    }

    float bestd[8]; int besti[8];
    #pragma unroll
    for (int r = 0; r < 8; ++r) { bestd[r] = 3.4e38f; besti[r] = 0; }
    #pragma unroll
    for (int ch = 0; ch < NCHUNK; ++ch) {
        v8f cz = {0.f, 0.f, 0.f, 0.f, 0.f, 0.f, 0.f, 0.f};
        v8f d = __builtin_amdgcn_wmma_f32_16x16x4_f32(false, a, false, bmat[ch],
                                                      (short)0, cz, false, false);
        int n = ch * 16 + (lane & 15);
        #pragma unroll
        for (int r = 0; r < 8; ++r) {
            float dv = d[r];
            // strict < keeps earlier (lower-index) chunk on ties
            bool lt = dv < bestd[r];
            bestd[r] = lt ? dv : bestd[r];
            besti[r] = lt ? n  : besti[r];
        }
    }

    // Branchless cross-lane argmin: pack (sortable dist, idx) into u64, min-reduce
    // over the 16 lanes of each D row (xor net stays inside each half-wave group).
    unsigned long long keys[8];
    #pragma unroll
    for (int r = 0; r < 8; ++r)
        keys[r] = ((unsigned long long)sortable_f32(bestd[r]) << 32) | (unsigned)besti[r];
    #pragma unroll
    for (int r = 0; r < 8; ++r) {
        #pragma unroll
        for (int off = 1; off < 16; off <<= 1) {
            unsigned long long o = __shfl_xor(keys[r], off, 32);
            keys[r] = (o < keys[r]) ? o : keys[r];
        }
    }
    // row r holds pixel M=r (lanes 0-15) and M=r+8 (lanes 16-31); route to lane==pixel
    int closest = 0;
    #pragma unroll
    for (int r = 0; r < 8; ++r) {
        int lo = (int)(unsigned)__shfl(keys[r], 0, 32);
        int hi = (int)(unsigned)__shfl(keys[r], 16, 32);
        closest = (lane == r)     ? lo : closest;
        closest = (lane == r + 8) ? hi : closest;
    }
    if (lane < 16) {
        int s = sem[b * HW + p];
        int thing = (THING_MASK >> s) & 1;
        int any = meta[b];
        inst_out[b * HW + p] = (thing && any) ? (closest + 1) : 0;
    }
}

// ---------------- kernel 5: (instance,class) histogram + stuff counts ----------------
__global__ void k_hist(const int* __restrict__ sem, const int* __restrict__ inst,
                       int* hist, int* scnt) {
    __shared__ int s_h[(KK + 1) * NC];
    __shared__ int s_s[NC];
    int tid = threadIdx.x;
    for (int i = tid; i < (KK + 1) * NC; i += 256) s_h[i] = 0;
    if (tid < NC) s_s[tid] = 0;
    __syncthreads();
    int base = blockIdx.x * 2048;   // 2048 | HW -> no batch straddle
    int b = base / HW;
    for (int i = tid; i < 2048; i += 256) {
        int g = base + i;
        int s = sem[g], in = inst[g];
        atomicAdd(&s_h[in * NC + s], 1);
        if (in == 0) atomicAdd(&s_s[s], 1);
    }
    __syncthreads();
    int* gh = hist + b * (KK + 1) * NC;
    for (int i = tid; i < (KK + 1) * NC; i += 256)
        if (s_h[i]) atomicAdd(&gh[i], s_h[i]);
    if (tid < NC && s_s[tid]) atomicAdd(&scnt[b * NC + tid], s_s[tid]);
}

// ---------------- kernel 6: per-instance majority class ----------------
__global__ void k_major(const int* __restrict__ hist, int* __restrict__ maj) {
    int t = blockIdx.x * blockDim.x + threadIdx.x;   // over B*(K+1)
    if (t >= BB * (KK + 1)) return;
    const int* h = hist + t * NC;
    int best = h[0], bi = 0;
    #pragma unroll
    for (int c = 1; c < NC; ++c) if (h[c] > best) { best = h[c]; bi = c; }
    maj[t] = bi;
}

// ---------------- kernel 7: final panoptic labels ----------------
__global__ void k_final(const int* __restrict__ sem, const int* __restrict__ inst,
                        const int* __restrict__ maj, const int* __restrict__ scnt,
                        int* __restrict__ pan) {
    int g = blockIdx.x * blockDim.x + threadIdx.x;
    if (g >= BB * HW) return;
    int b = g / HW;
    int s = sem[g], in = inst[g];
    int out;
    if (in > 0) {
        out = maj[b * (KK + 1) + in] * 256 + in;
    } else {
        out = VOIDL * 256;
        int thing = (THING_MASK >> s) & 1;
        if (!thing && scnt[b * NC + s] >= STUFF_AREA) out = s * 256;
    }
    pan[g] = out;
}

extern "C" void kernel_launch(void* const* d_in, const int* in_sizes, int n_in,
                              void* d_out, int out_size, void* d_ws, size_t ws_size,
                              hipStream_t stream) {
    const float* logits = (const float*)d_in[0];   // [B,19,H,W]
    const float* heat   = (const float*)d_in[1];   // [B,1,H,W]
    const float* offm   = (const float*)d_in[2];   // [B,2,H,W]
    int* out  = (int*)d_out;                       // int32 outputs
    int* pan  = out;                               // [B,H,W]
    int* inst = out + BB * HW;                     // [B,H,W]

    char* ws = (char*)d_ws;
    int*   sem  = (int*)ws;                                       // B*HW ints
    float* psc  = (float*)(ws + (size_t)4 * BB * HW);             // B*PEAK_CAP
    int*   pix  = (int*)((char*)psc + (size_t)4 * BB * PEAK_CAP); // B*PEAK_CAP
    int*   pcnt = (int*)((char*)pix + (size_t)4 * BB * PEAK_CAP); // B (padded 256B)
    float* cy   = (float*)((char*)pcnt + 256);                    // B*KPAD
    float* cx   = (float*)((char*)cy + 4 * BB * KPAD);            // B*KPAD
    int*   meta = (int*)((char*)cx + 4 * BB * KPAD);              // B (padded 256B)
    int*   hist = (int*)((char*)meta + 256);                      // B*(K+1)*NC
    int*   scnt = (int*)((char*)hist + 4 * BB * (KK + 1) * NC);   // B*NC (padded)
    int*   maj  = (int*)((char*)scnt + 256);                      // B*(K+1)

    k_zero<<<30, 256, 0, stream>>>(pcnt, meta, hist, scnt);
    k_sem_argmax<<<(BB * HW + 255) / 256, 256, 0, stream>>>(logits, sem);
    k_nms<<<dim3(WW / 16, HH / 16, BB), dim3(16, 16), 0, stream>>>(heat, psc, pix, pcnt);
    k_topk<<<BB, 256, 0, stream>>>(psc, pix, pcnt, cy, cx, meta);
    k_closest<<<BB * HW / 128, 256, 0, stream>>>(offm, sem, cy, cx, meta, inst);
    k_hist<<<BB * HW / 2048, 256, 0, stream>>>(sem, inst, hist, scnt);
    k_major<<<(BB * (KK + 1) + 255) / 256, 256, 0, stream>>>(hist, maj);
    k_final<<<(BB * HW + 255) / 256, 256, 0, stream>>>(sem, inst, maj, scnt, pan);
}